// QuantizerUniformLayer_78975858639646
// MI455X (gfx1250) — compile-verified
//
#include <hip/hip_runtime.h>
#include <stdint.h>

// ---------------------------------------------------------------------------
// Nearest-codeword quantization, uniform sorted 16-entry codebook.
// Memory-bound streaming kernel: 512 MiB total traffic -> ~23 us @ 23.3 TB/s.
// No matmul structure => WMMA not applicable; we exercise the CDNA5 async
// GLOBAL_LOAD_ASYNC_TO_LDS_B128 / ASYNCcnt pipeline + non-temporal stores.
// ---------------------------------------------------------------------------

#define TPB            256
#define NCHUNK         4                         // 4 x B128 per lane per tile
#define TILE_FLOATS    4096                      // 256 lanes * 16 floats = 16 KB
#define TILE_BYTES     16384
#define TILES_PER_BLK  8
#define NBLOCKS        2048                      // 2048*8*4096 = 64Mi floats exactly

typedef __attribute__((ext_vector_type(4))) float f32x4;
typedef __attribute__((ext_vector_type(4))) int   i32x4;
typedef __attribute__((address_space(1))) f32x4       gf32x4;
typedef const __attribute__((address_space(1))) f32x4 cgf32x4;
typedef __attribute__((address_space(1))) i32x4*      g_v4i_p;  // builtin arg0 type
typedef __attribute__((address_space(3))) i32x4*      l_v4i_p;  // builtin arg1 type

#if defined(__HIP_DEVICE_COMPILE__) && defined(__gfx1250__) && \
    __has_builtin(__builtin_amdgcn_global_load_async_to_lds_b128)
#define USE_ASYNC 1
#else
#define USE_ASYNC 0
#endif

#if defined(__HIP_DEVICE_COMPILE__) && __has_builtin(__builtin_amdgcn_s_wait_asynccnt)
#define WAIT_ASYNC(n) do { __builtin_amdgcn_s_wait_asynccnt(n); \
                           asm volatile("" ::: "memory"); } while (0)
#else
#define WAIT_ASYNC(n) asm volatile("s_wait_asynccnt " #n ::: "memory")
#endif

// Analytic nearest-codeword for a uniform sorted codebook c_k = c0 + k*step.
// Matches the reference exactly: bracket with floor, compare distances,
// tie-break toward the LOWER codeword (<=), clamp to [0,15].
__device__ __forceinline__ float quantize1(float v, float c0, float step, float inv) {
    float t   = (v - c0) * inv;
    float fl  = __builtin_floorf(t);
    float lo  = __builtin_fminf(__builtin_fmaxf(fl, 0.0f), 15.0f);
    float hi  = __builtin_fminf(lo + 1.0f, 15.0f);
    float clo = __builtin_fmaf(lo, step, c0);
    float chi = __builtin_fmaf(hi, step, c0);
    return (__builtin_fabsf(v - clo) <= __builtin_fabsf(v - chi)) ? clo : chi;
}

__device__ __forceinline__ f32x4 quantize4(f32x4 v, float c0, float step, float inv) {
    f32x4 r;
    r.x = quantize1(v.x, c0, step, inv);
    r.y = quantize1(v.y, c0, step, inv);
    r.z = quantize1(v.z, c0, step, inv);
    r.w = quantize1(v.w, c0, step, inv);
    return r;
}

__global__ void __launch_bounds__(TPB)
quantizer_uniform_kernel(const float* __restrict__ xin,
                         const float* __restrict__ cb,
                         float* __restrict__ xout) {
    const int  t         = threadIdx.x;
    const long blockBase = (long)blockIdx.x * (TILES_PER_BLK * TILE_FLOATS);

    // Uniform codebook parameters (scalar loads; uniform across the wave).
    const float c0   = cb[0];
    const float cK   = cb[15];
    const float step = (cK - c0) * (1.0f / 15.0f);
    const float inv  = 1.0f / step;

    gf32x4* outg = (gf32x4*)(xout + blockBase);

#if USE_ASYNC
    __shared__ f32x4 smem[2 * (TILE_FLOATS / 4)];   // 32 KB, double-buffered

    const __attribute__((address_space(1))) char* gsrc =
        (const __attribute__((address_space(1))) char*)(xin + blockBase);
    __attribute__((address_space(3))) char* lds =
        (__attribute__((address_space(3))) char*)smem;

    // Prologue: tile 0 -> buffer 0 (4 async B128 loads per lane; no VGPR dest).
    #pragma unroll
    for (int k = 0; k < NCHUNK; ++k) {
        __builtin_amdgcn_global_load_async_to_lds_b128(
            (g_v4i_p)(gsrc + (size_t)k * 4096 + (size_t)t * 16),
            (l_v4i_p)(lds  + k * 4096 + t * 16),
            0, 0);
    }

    for (int tile = 0; tile < TILES_PER_BLK; ++tile) {
        const int buf = tile & 1;

        if (tile + 1 < TILES_PER_BLK) {
            // Run ahead: issue next tile's loads into the other buffer.
            const __attribute__((address_space(1))) char* gnext =
                gsrc + (size_t)(tile + 1) * TILE_BYTES;
            __attribute__((address_space(3))) char* lnext =
                lds + (buf ^ 1) * TILE_BYTES;
            #pragma unroll
            for (int k = 0; k < NCHUNK; ++k) {
                __builtin_amdgcn_global_load_async_to_lds_b128(
                    (g_v4i_p)(gnext + (size_t)k * 4096 + (size_t)t * 16),
                    (l_v4i_p)(lnext + k * 4096 + t * 16),
                    0, 0);
            }
            WAIT_ASYNC(4);   // current tile (first 4 of 8 outstanding) complete
        } else {
            WAIT_ASYNC(0);
        }

        // Each lane consumes exactly the LDS bytes it async-loaded: no barrier.
        const f32x4* sbuf = (const f32x4*)(smem + buf * (TILE_FLOATS / 4));
        gf32x4*      obas = outg + (long)tile * (TILE_FLOATS / 4);
        #pragma unroll
        for (int k = 0; k < NCHUNK; ++k) {
            f32x4 v = sbuf[k * TPB + t];                       // ds_load_b128
            f32x4 r = quantize4(v, c0, step, inv);
            __builtin_nontemporal_store(r, obas + k * TPB + t); // b128, TH=NT
        }
    }
#else
    // Fallback: direct non-temporal B128 streaming (same tiling).
    cgf32x4* ing = (cgf32x4*)(xin + blockBase);
    for (int tile = 0; tile < TILES_PER_BLK; ++tile) {
        #pragma unroll
        for (int k = 0; k < NCHUNK; ++k) {
            const long i = (long)tile * (TILE_FLOATS / 4) + k * TPB + t;
            f32x4 v = __builtin_nontemporal_load(ing + i);
            f32x4 r = quantize4(v, c0, step, inv);
            __builtin_nontemporal_store(r, outg + i);
        }
    }
#endif
}

extern "C" void kernel_launch(void* const* d_in, const int* in_sizes, int n_in,
                              void* d_out, int out_size, void* d_ws, size_t ws_size,
                              hipStream_t stream) {
    const float* x  = (const float*)d_in[0];   // 8192*8192 fp32
    const float* cb = (const float*)d_in[1];   // 16 fp32, uniform sorted
    float* out      = (float*)d_out;
    (void)in_sizes; (void)n_in; (void)out_size; (void)d_ws; (void)ws_size;

    quantizer_uniform_kernel<<<NBLOCKS, TPB, 0, stream>>>(x, cb, out);
}